// GNNStack_65678639890511
// MI455X (gfx1250) — compile-verified
//
#include <hip/hip_runtime.h>
#include <hip/hip_bf16.h>

// ---------------- problem constants ----------------
#define Nn 100000
#define Ee 1600000
#define Dd 128
#define Hh 128
#define Cc 10
#define Gg 64

typedef __attribute__((ext_vector_type(2))) float v2f;
typedef __attribute__((ext_vector_type(8))) float v8f;

// ---------------- elementwise copy (float4) ----------------
__global__ __launch_bounds__(256) void copy_rows(const float* __restrict__ src,
                                                 float* __restrict__ dst) {
    // N*32 threads, each moves one float4 (row-major [N,128])
    long long t = (long long)blockIdx.x * blockDim.x + threadIdx.x;
    if (t >= (long long)Nn * 32) return;
    ((float4*)dst)[t] = ((const float4*)src)[t];
}

// ---------------- edge scatter-add: acc[dst] += x[src] ----------------
__global__ __launch_bounds__(256) void edge_scatter(const int* __restrict__ src,
                                                    const int* __restrict__ dst,
                                                    const float* __restrict__ x,
                                                    float* __restrict__ acc) {
    // one wave per edge; lane handles 4 consecutive features
    long long t = (long long)blockIdx.x * blockDim.x + threadIdx.x;
    long long e = t >> 5;
    if (e >= Ee) return;
    int lane = (int)(t & 31);
    int sn = src[e];
    int dn = dst[e];
    float4 v = ((const float4*)(x + (long long)sn * Hh))[lane];
    float* p = acc + (long long)dn * Hh + lane * 4;
    atomicAdd(p + 0, v.x);
    atomicAdd(p + 1, v.y);
    atomicAdd(p + 2, v.z);
    atomicAdd(p + 3, v.w);
}

// ---------------- fused GIN MLP: out = relu(in@w1+b1)@w2+b2 ----------------
// block = 256 threads (8 wave32), one block handles 16 node rows.
// wave w computes the 16-wide N-tile [16w, 16w+16) with V_WMMA_F32_16X16X4_F32.
#define LDSS 132   // padded row stride for the 16x128 intermediate
__global__ __launch_bounds__(256) void gin_mlp(const float* __restrict__ in,
                                               const float* __restrict__ w1,
                                               const float* __restrict__ b1,
                                               const float* __restrict__ w2,
                                               const float* __restrict__ b2,
                                               float* __restrict__ out) {
    __shared__ float mid[16 * LDSS];
    const int r0   = blockIdx.x * 16;           // node-row base (N divisible by 16)
    const int lane = threadIdx.x & 31;
    const int lr   = lane & 15;                 // row/col within half-wave
    const int hi   = lane >> 4;                 // 0 = lanes 0-15, 1 = lanes 16-31
    const int koff = hi * 2;                    // K sub-offset per ISA A/B layout
    const int n0   = (threadIdx.x >> 5) * 16;   // N-tile base owned by this wave

    // ---- GEMM1: A = in[r0+lr, :]  B = w1[:, n0..n0+15] ----
    v8f acc = {};
    const float* arow = in + (long long)(r0 + lr) * Hh + koff;
    for (int k = 0; k < Hh; k += 4) {
        v2f a = *(const v2f*)(arow + k);        // K = k+koff, k+koff+1
        v2f b;
        b.x = w1[(k + koff) * Hh + n0 + lr];
        b.y = w1[(k + koff + 1) * Hh + n0 + lr];
        acc = __builtin_amdgcn_wmma_f32_16x16x4_f32(false, a, false, b,
                                                    (short)0, acc, false, false);
    }
    float bias1 = b1[n0 + lr];
    #pragma unroll
    for (int v = 0; v < 8; ++v) {
        int m = v + hi * 8;                     // D layout: M = v (+8 for upper half)
        float val = acc[v] + bias1;
        mid[m * LDSS + n0 + lr] = val > 0.f ? val : 0.f;
    }
    __syncthreads();

    // ---- GEMM2: A = mid[lr, :] (LDS)  B = w2[:, n0..n0+15] ----
    v8f acc2 = {};
    for (int k = 0; k < Hh; k += 4) {
        v2f a;
        a.x = mid[lr * LDSS + k + koff];
        a.y = mid[lr * LDSS + k + koff + 1];
        v2f b;
        b.x = w2[(k + koff) * Hh + n0 + lr];
        b.y = w2[(k + koff + 1) * Hh + n0 + lr];
        acc2 = __builtin_amdgcn_wmma_f32_16x16x4_f32(false, a, false, b,
                                                     (short)0, acc2, false, false);
    }
    float bias2 = b2[n0 + lr];
    #pragma unroll
    for (int v = 0; v < 8; ++v) {
        int m = v + hi * 8;
        out[(long long)(r0 + m) * Hh + n0 + lr] = acc2[v] + bias2;
    }
}

// ---------------- relu + layernorm, one wave per row, in-place safe ----------------
__global__ __launch_bounds__(256) void relu_ln(const float* __restrict__ in,
                                               const float* __restrict__ g,
                                               const float* __restrict__ b,
                                               float* __restrict__ out) {
    long long t = (long long)blockIdx.x * blockDim.x + threadIdx.x;
    long long row = t >> 5;
    if (row >= Nn) return;
    int lane = (int)(t & 31);
    float4 v = ((const float4*)(in + row * Hh))[lane];
    v.x = v.x > 0.f ? v.x : 0.f;
    v.y = v.y > 0.f ? v.y : 0.f;
    v.z = v.z > 0.f ? v.z : 0.f;
    v.w = v.w > 0.f ? v.w : 0.f;
    float s  = v.x + v.y + v.z + v.w;
    float sq = v.x * v.x + v.y * v.y + v.z * v.z + v.w * v.w;
    #pragma unroll
    for (int off = 16; off > 0; off >>= 1) {
        s  += __shfl_xor(s, off, 32);
        sq += __shfl_xor(sq, off, 32);
    }
    float mu  = s * (1.0f / Hh);
    float var = sq * (1.0f / Hh) - mu * mu;
    float rstd = rsqrtf(var + 1e-5f);
    float4 gv = ((const float4*)g)[lane];
    float4 bv = ((const float4*)b)[lane];
    float4 o;
    o.x = (v.x - mu) * rstd * gv.x + bv.x;
    o.y = (v.y - mu) * rstd * gv.y + bv.y;
    o.z = (v.z - mu) * rstd * gv.z + bv.z;
    o.w = (v.w - mu) * rstd * gv.w + bv.w;
    ((float4*)(out + row * Hh))[lane] = o;
}

// ---------------- plain relu (layer 2: emb must stay pre-relu) ----------------
__global__ __launch_bounds__(256) void relu_only(const float* __restrict__ in,
                                                 float* __restrict__ out) {
    long long t = (long long)blockIdx.x * blockDim.x + threadIdx.x;
    if (t >= (long long)Nn * 32) return;
    float4 v = ((const float4*)in)[t];
    v.x = v.x > 0.f ? v.x : 0.f;
    v.y = v.y > 0.f ? v.y : 0.f;
    v.z = v.z > 0.f ? v.z : 0.f;
    v.w = v.w > 0.f ? v.w : 0.f;
    ((float4*)out)[t] = v;
}

// ---------------- zero small pooling scratch ----------------
__global__ __launch_bounds__(256) void zero_buf(float* __restrict__ p, int n) {
    for (int i = threadIdx.x; i < n; i += blockDim.x) p[i] = 0.f;
}

// ---------------- mean-pool accumulation (atomics) ----------------
__global__ __launch_bounds__(256) void pool_acc(const float* __restrict__ h,
                                                const int* __restrict__ batch,
                                                float* __restrict__ sums,
                                                float* __restrict__ cnts) {
    long long t = (long long)blockIdx.x * blockDim.x + threadIdx.x;
    long long node = t >> 5;
    if (node >= Nn) return;
    int lane = (int)(t & 31);
    int g = batch[node];
    float4 v = ((const float4*)(h + node * Hh))[lane];
    float* p = sums + (long long)g * Hh + lane * 4;
    atomicAdd(p + 0, v.x);
    atomicAdd(p + 1, v.y);
    atomicAdd(p + 2, v.z);
    atomicAdd(p + 3, v.w);
    if (lane == 0) atomicAdd(cnts + g, 1.0f);
}

// ---------------- post_mp: pooled@pw1+pb1 -> @pw2+pb2 -> log_softmax ----------------
__global__ __launch_bounds__(256) void post_mp(const float* __restrict__ sums,
                                               const float* __restrict__ cnts,
                                               const float* __restrict__ pw1,
                                               const float* __restrict__ pb1,
                                               const float* __restrict__ pw2,
                                               const float* __restrict__ pb2,
                                               float* __restrict__ outls) {
    __shared__ float pooled[Gg * Hh];
    int t = threadIdx.x;
    for (int i = t; i < Gg * Hh; i += 256) {
        int g = i >> 7;
        float c = cnts[g];
        c = c > 1.f ? c : 1.f;
        pooled[i] = sums[i] / c;
    }
    __syncthreads();
    if (t < Gg) {
        float lg[Cc];
        #pragma unroll
        for (int c = 0; c < Cc; ++c) lg[c] = pb2[c];
        for (int j = 0; j < Hh; ++j) {
            float zj = pb1[j];
            for (int h = 0; h < Hh; ++h) zj += pooled[t * Hh + h] * pw1[h * Hh + j];
            #pragma unroll
            for (int c = 0; c < Cc; ++c) lg[c] += zj * pw2[j * Cc + c];
        }
        float m = lg[0];
        #pragma unroll
        for (int c = 1; c < Cc; ++c) m = fmaxf(m, lg[c]);
        float s = 0.f;
        #pragma unroll
        for (int c = 0; c < Cc; ++c) s += __expf(lg[c] - m);
        float lse = __logf(s);
        #pragma unroll
        for (int c = 0; c < Cc; ++c) outls[t * Cc + c] = lg[c] - m - lse;
    }
}

extern "C" void kernel_launch(void* const* d_in, const int* in_sizes, int n_in,
                              void* d_out, int out_size, void* d_ws, size_t ws_size,
                              hipStream_t stream) {
    (void)in_sizes; (void)n_in; (void)out_size; (void)ws_size;

    const float* x     = (const float*)d_in[0];
    const int*   ei    = (const int*)d_in[1];
    const int*   src   = ei;
    const int*   dst   = ei + Ee;
    const int*   batch = (const int*)d_in[2];
    const float* w1[3] = {(const float*)d_in[3], (const float*)d_in[7],  (const float*)d_in[11]};
    const float* b1[3] = {(const float*)d_in[4], (const float*)d_in[8],  (const float*)d_in[12]};
    const float* w2[3] = {(const float*)d_in[5], (const float*)d_in[9],  (const float*)d_in[13]};
    const float* b2[3] = {(const float*)d_in[6], (const float*)d_in[10], (const float*)d_in[14]};
    const float* lng[2] = {(const float*)d_in[15], (const float*)d_in[17]};
    const float* lnb[2] = {(const float*)d_in[16], (const float*)d_in[18]};
    const float* pw1 = (const float*)d_in[19];
    const float* pb1 = (const float*)d_in[20];
    const float* pw2 = (const float*)d_in[21];
    const float* pb2 = (const float*)d_in[22];

    float* W0   = (float*)d_ws;                         // scatter accumulator / MLP input
    float* W1   = W0 + (size_t)Nn * Hh;                 // MLP output / LN output / relu(h)
    float* sums = W1 + (size_t)Nn * Hh;                 // [G,H]
    float* cnts = sums + Gg * Hh;                       // [G]
    float* emb   = (float*)d_out;                       // [N,H] output region
    float* outls = emb + (size_t)Nn * Hh;               // [G,C] output region

    const int copyBlocks    = (Nn * 32 + 255) / 256;    // 12500
    const int scatterBlocks = (int)(((long long)Ee * 32 + 255) / 256); // 200000
    const int mlpBlocks     = Nn / 16;                  // 6250

    const float* cur = x;
    for (int i = 0; i < 3; ++i) {
        // s = cur + segment_sum(cur[src], dst)
        copy_rows<<<copyBlocks, 256, 0, stream>>>(cur, W0);
        edge_scatter<<<scatterBlocks, 256, 0, stream>>>(src, dst, cur, W0);
        // conv output
        float* o = (i == 2) ? emb : W1;
        gin_mlp<<<mlpBlocks, 256, 0, stream>>>(W0, w1[i], b1[i], w2[i], b2[i], o);
        if (i < 2) {
            relu_ln<<<copyBlocks, 256, 0, stream>>>(W1, lng[i], lnb[i], W1);
            cur = W1;
        } else {
            relu_only<<<copyBlocks, 256, 0, stream>>>(emb, W1);
        }
    }

    zero_buf<<<1, 256, 0, stream>>>(sums, Gg * Hh + Gg);
    pool_acc<<<copyBlocks, 256, 0, stream>>>(W1, batch, sums, cnts);
    post_mp<<<1, 256, 0, stream>>>(sums, cnts, pw1, pb1, pw2, pb2, outls);
}